// MambaBlock_87359634801396
// MI455X (gfx1250) — compile-verified
//
#include <hip/hip_runtime.h>
#include <hip/hip_bf16.h>

typedef __attribute__((ext_vector_type(2))) float v2f;
typedef __attribute__((ext_vector_type(8))) float v8f;
typedef __attribute__((ext_vector_type(4))) int v4i;
typedef __attribute__((address_space(1))) v4i gv4i;  // global int4
typedef __attribute__((address_space(3))) v4i lv4i;  // LDS int4

constexpr int Bn = 8, S = 4096, D = 1024, N = 16;

// ---------------------------------------------------------------------------
// K1: CW[n][d] = sum_k C[n][k] * W_out[d][k]   (16 x 1024, K=1024) — tiny GEMM
// grid 64 x 256 threads: one thread per (n, d)
// ---------------------------------------------------------------------------
__global__ void k_cw(const float* __restrict__ C, const float* __restrict__ W,
                     float* __restrict__ CW) {
  int tid = threadIdx.x;
  int n = tid & 15;
  int d = blockIdx.x * 16 + (tid >> 4);
  const float4* wr = reinterpret_cast<const float4*>(W + (size_t)d * D);
  const float4* cr = reinterpret_cast<const float4*>(C + (size_t)n * D);
  float a0 = 0.f, a1 = 0.f, a2 = 0.f, a3 = 0.f;
#pragma unroll 4
  for (int i = 0; i < D / 4; ++i) {
    float4 w = wr[i], c = cr[i];
    a0 += w.x * c.x; a1 += w.y * c.y; a2 += w.z * c.z; a3 += w.w * c.w;
  }
  CW[(size_t)n * D + d] = (a0 + a1) + (a2 + a3);
}

// ---------------------------------------------------------------------------
// K2: xb[r][n] = sum_k x_row(r)[k] * B[k][n], r = s*8 + b  (32768 x 16, K=1024)
// fp32 WMMA 16x16x4, dual accumulators to halve the WMMA RAW chain.
// B staged in LDS via async-to-LDS DMA (ASYNCcnt path) when available.
// grid 256 x 256 threads (8 waves/block, 2048 tiles total)
// ---------------------------------------------------------------------------
__global__ void k_xb(const float* __restrict__ x, const float* __restrict__ Bm,
                     float* __restrict__ xb) {
  __shared__ float Blds[D * N];  // 64 KB

#if __has_builtin(__builtin_amdgcn_global_load_async_to_lds_b128)
  {
    gv4i* gsrc = (gv4i*)Bm;       // C-style cast: const-drop + addrspacecast
    lv4i* ldst = (lv4i*)Blds;
    for (int i = threadIdx.x; i < (D * N) / 4; i += blockDim.x)
      __builtin_amdgcn_global_load_async_to_lds_b128(gsrc + i, ldst + i, 0, 0);
#if __has_builtin(__builtin_amdgcn_s_wait_asynccnt)
    __builtin_amdgcn_s_wait_asynccnt(0);
#else
    asm volatile("s_wait_asynccnt 0x0" ::: "memory");
#endif
  }
#else
#warning "no __builtin_amdgcn_global_load_async_to_lds_b128; using VALU staging"
  for (int i = threadIdx.x; i < (D * N) / 4; i += blockDim.x)
    reinterpret_cast<float4*>(Blds)[i] = reinterpret_cast<const float4*>(Bm)[i];
#endif
  __syncthreads();

  int lane = threadIdx.x & 31;
  int tile = blockIdx.x * 8 + (threadIdx.x >> 5);  // 0..2047
  int r0 = tile * 16;
  int m = lane & 15;                 // A row within tile / B column n
  int off = (lane >> 4) << 1;        // 0 for lanes 0-15, 2 for lanes 16-31
  int r = r0 + m;                    // r = s*8 + b
  const float* xrow = x + ((size_t)(r & 7) * S + (size_t)(r >> 3)) * D;

  v8f c0 = {0.f, 0.f, 0.f, 0.f, 0.f, 0.f, 0.f, 0.f};
  v8f c1 = {0.f, 0.f, 0.f, 0.f, 0.f, 0.f, 0.f, 0.f};
#pragma unroll 4
  for (int j = 0; j < D / 4; j += 2) {
    v2f a0 = *reinterpret_cast<const v2f*>(xrow + 4 * j + off);
    v2f a1 = *reinterpret_cast<const v2f*>(xrow + 4 * (j + 1) + off);
    v2f b0, b1;
    b0.x = Blds[(4 * j + off) * N + m];
    b0.y = Blds[(4 * j + off + 1) * N + m];
    b1.x = Blds[(4 * (j + 1) + off) * N + m];
    b1.y = Blds[(4 * (j + 1) + off + 1) * N + m];
    c0 = __builtin_amdgcn_wmma_f32_16x16x4_f32(false, a0, false, b0, (short)0,
                                               c0, false, false);
    c1 = __builtin_amdgcn_wmma_f32_16x16x4_f32(false, a1, false, b1, (short)0,
                                               c1, false, false);
  }
  v8f c = c0 + c1;
#pragma unroll
  for (int v = 0; v < 8; ++v) {
    int M = v + ((lane >> 4) << 3);
    xb[(size_t)(r0 + M) * N + m] = c[v];
  }
}

// ---------------------------------------------------------------------------
// K3: sequential scan  h_t = tanh(xb_t + A h_{t-1})  (8 independent batches)
// 1 block x 128 threads: lane = b*16 + n inside each wave (2 batches / wave32)
// Broadcast of h within each 16-lane batch group via v_permlane16_b32
// (pure VALU, no LDS round trip) when available; ds_bpermute fallback.
// ---------------------------------------------------------------------------
__global__ void k_scan(const float* __restrict__ xb, const float* __restrict__ A,
                       float* __restrict__ hs) {
  int tid = threadIdx.x;   // 0..127 = b*16 + n
  int lane = tid & 31;
  int n = tid & 15;

  float Ar[16];
#pragma unroll
  for (int k = 0; k < 16; ++k) Ar[k] = A[n * 16 + k];

  const int base4 = (lane & 16) << 2;  // byte base for bpermute within group
  float h = 0.f;
  float xc = xb[tid];
  for (int t = 0; t < S; ++t) {
    int t1 = (t + 1 < S) ? t + 1 : t;
    float xn = xb[(size_t)t1 * 128 + tid];  // prefetch next step's input

    int hb = __float_as_int(h);
    float hv[16];
#pragma unroll
    for (int k = 0; k < 16; ++k) {
#if __has_builtin(__builtin_amdgcn_permlane16)
      // broadcast lane k within each 16-lane half: all selector nibbles = k
      unsigned sel = 0x11111111u * (unsigned)k;
      hv[k] = __int_as_float(
          __builtin_amdgcn_permlane16(hb, hb, sel, sel, false, false));
#else
      hv[k] = __int_as_float(
          __builtin_amdgcn_ds_bpermute(base4 + (k << 2), hb));
#endif
    }

    float s0 = xc, s1 = 0.f, s2 = 0.f, s3 = 0.f;
#pragma unroll
    for (int k = 0; k < 16; k += 4) {
      s0 += Ar[k] * hv[k];
      s1 += Ar[k + 1] * hv[k + 1];
      s2 += Ar[k + 2] * hv[k + 2];
      s3 += Ar[k + 3] * hv[k + 3];
    }
    float v = (s0 + s1) + (s2 + s3);
#if __has_builtin(__builtin_amdgcn_tanhf)
    h = __builtin_amdgcn_tanhf(v);
#else
    h = tanhf(v);
#endif
    hs[(size_t)t * 128 + tid] = h;
    xc = xn;
  }
}

// ---------------------------------------------------------------------------
// K4: out[b][s][d] = sum_n hs[s,b,n] * CW[n][d] + b_out[d]  (32768 x 1024, K=16)
// fp32 WMMA 16x16x4 (4 per tile); A loaded once per 16-row block, 64 d-tiles.
// grid 256 x 256 threads (8 waves/block, 2048 row-blocks)
// ---------------------------------------------------------------------------
__global__ void k_out(const float* __restrict__ hs, const float* __restrict__ CW,
                      const float* __restrict__ bout, float* __restrict__ out) {
  int lane = threadIdx.x & 31;
  int rb = blockIdx.x * 8 + (threadIdx.x >> 5);  // 0..2047
  int b = rb >> 8;                 // batch
  int s0 = (rb & 255) << 4;        // 16 consecutive seq positions
  int m = lane & 15;
  int off = (lane >> 4) << 1;

  v2f a[4];
  const float* hrow = hs + ((size_t)(s0 + m) * Bn + b) * N;
#pragma unroll
  for (int j = 0; j < 4; ++j)
    a[j] = *reinterpret_cast<const v2f*>(hrow + 4 * j + off);

  size_t obase = ((size_t)b * S + s0) * D;
  for (int dt = 0; dt < D / N; ++dt) {
    int d0 = dt * 16;
    float bias = bout[d0 + m];
    v8f c = {bias, bias, bias, bias, bias, bias, bias, bias};
#pragma unroll
    for (int j = 0; j < 4; ++j) {
      v2f bw;
      bw.x = CW[(size_t)(4 * j + off) * D + d0 + m];
      bw.y = CW[(size_t)(4 * j + off + 1) * D + d0 + m];
      c = __builtin_amdgcn_wmma_f32_16x16x4_f32(false, a[j], false, bw, (short)0,
                                                c, false, false);
    }
#pragma unroll
    for (int v = 0; v < 8; ++v) {
      int M = v + ((lane >> 4) << 3);
      out[obase + (size_t)M * D + d0 + m] = c[v];
    }
  }
}

// ---------------------------------------------------------------------------
extern "C" void kernel_launch(void* const* d_in, const int* in_sizes, int n_in,
                              void* d_out, int out_size, void* d_ws, size_t ws_size,
                              hipStream_t stream) {
  const float* x  = (const float*)d_in[0];
  const float* A  = (const float*)d_in[1];
  const float* Bm = (const float*)d_in[2];
  const float* C  = (const float*)d_in[3];
  const float* W  = (const float*)d_in[4];
  const float* bo = (const float*)d_in[5];
  float* out = (float*)d_out;

  float* ws = (float*)d_ws;
  float* xb = ws;                                   // 4096*8*16  = 524288 f
  float* hs = ws + (size_t)S * Bn * N;              // 524288 f
  float* cw = hs + (size_t)S * Bn * N;              // 16*1024 = 16384 f
                                                    // total ~4.1 MB scratch

  hipLaunchKernelGGL(k_cw,   dim3(64),  dim3(256), 0, stream, C, W, cw);
  hipLaunchKernelGGL(k_xb,   dim3(256), dim3(256), 0, stream, x, Bm, xb);
  hipLaunchKernelGGL(k_scan, dim3(1),   dim3(128), 0, stream, xb, A, hs);
  hipLaunchKernelGGL(k_out,  dim3(256), dim3(256), 0, stream, hs, cw, bo, out);
}